// EdgeCrossingLoss_66400194396557
// MI455X (gfx1250) — compile-verified
//
#include <hip/hip_runtime.h>
#include <hip/hip_bf16.h>

typedef float v2f __attribute__((ext_vector_type(2)));
typedef float v8f __attribute__((ext_vector_type(8)));

#define KNB 19       // neighbors used (top-20 minus self)
#define TPI 4        // candidate tiles (of 16) per barrier period -> 64 cands
#define RSTRIDE 68   // LDS row stride in floats (16B aligned, low-conflict)

// ---------------------------------------------------------------------------
// Kernel 1: per-face centroid (x,y,z,|c|^2) and 3 edge vectors
// ---------------------------------------------------------------------------
__global__ __launch_bounds__(256)
void ecl_prep_kernel(const float* __restrict__ verts,
                     const int* __restrict__ faces,
                     float4* __restrict__ cent,
                     float4* __restrict__ edg,
                     int F) {
    int f = blockIdx.x * blockDim.x + threadIdx.x;
    if (f >= F) return;
    int i0 = faces[3 * f + 0];
    int i1 = faces[3 * f + 1];
    int i2 = faces[3 * f + 2];
    float ax = verts[3 * i0 + 0], ay = verts[3 * i0 + 1], az = verts[3 * i0 + 2];
    float bx = verts[3 * i1 + 0], by = verts[3 * i1 + 1], bz = verts[3 * i1 + 2];
    float cx = verts[3 * i2 + 0], cy = verts[3 * i2 + 1], cz = verts[3 * i2 + 2];
    float mx = (ax + bx + cx) / 3.0f;
    float my = (ay + by + cy) / 3.0f;
    float mz = (az + bz + cz) / 3.0f;
    float cn = mx * mx + my * my + mz * mz;
    cent[f] = make_float4(mx, my, mz, cn);
    // edge j = v[perm[j]] - v[j], perm = {1,2,0}
    edg[3 * f + 0] = make_float4(bx - ax, by - ay, bz - az, 0.0f);
    edg[3 * f + 1] = make_float4(cx - bx, cy - by, cz - bz, 0.0f);
    edg[3 * f + 2] = make_float4(ax - cx, ay - cy, az - cz, 0.0f);
}

// ---------------------------------------------------------------------------
// Kernel 2: WMMA-based k-NN. Each wave owns 32 query rows (two 16x4 A tiles);
// per barrier period it runs 8 v_wmma_f32_16x16x4_f32 over 64 candidates,
// stages s = |c_j|^2 - 2*q.c_j in LDS, then each lane scans its own row with
// float4 LDS reads and maintains a private sorted top-19 (strict '<' insert
// reproduces top_k's lower-index tie-break; skipping j==row reproduces the
// -inf diagonal / self-drop).
// ---------------------------------------------------------------------------
__global__ __launch_bounds__(128)
void ecl_topk_kernel(const float4* __restrict__ cent,
                     int* __restrict__ nn,
                     int F) {
    __shared__ __align__(16) float tile[4][32 * RSTRIDE];

    if ((int)(blockIdx.x * 128) >= F * 4) return;  // block-uniform guard

    const int lane = threadIdx.x & 31;
    const int wv   = threadIdx.x >> 5;
    const int lo   = lane & 15;          // row (A) / col (B) within tile
    const int h    = lane >> 4;          // lane half selects K pair
    const int qbase = (blockIdx.x * 4 + wv) * 32;

    // A tiles: 16x4 f32. Lanes 0-15 hold K={0,1}, lanes 16-31 hold K={2,3}.
    float4 cq0 = cent[qbase + lo];
    float4 cq1 = cent[qbase + 16 + lo];
    v2f a0, a1;
    a0.x = h ? cq0.z : cq0.x;  a0.y = h ? 0.0f : cq0.y;
    a1.x = h ? cq1.z : cq1.x;  a1.y = h ? 0.0f : cq1.y;

    float dist[KNB];
    int   idx[KNB];
#pragma unroll
    for (int m = 0; m < KNB; ++m) { dist[m] = __builtin_inff(); idx[m] = 0; }

    const int r = qbase + lane;          // query row owned by this lane
    const int nsuper = F >> 6;           // 64 candidates per period
    float* T = tile[wv];

    for (int sp = 0; sp < nsuper; ++sp) {
        const int cbase = sp * 64;
        if (sp + 1 < nsuper) {
            __builtin_prefetch(&cent[cbase + 64 + lane], 0, 3);
            __builtin_prefetch(&cent[cbase + 96 + lane], 0, 3);
        }
#pragma unroll
        for (int t = 0; t < TPI; ++t) {
            float4 cc = cent[cbase + t * 16 + lo];
            // B tile: 4x16 f32. VGPR0: K = h ; VGPR1: K = 2+h (K=3 zero pad).
            v2f b;
            b.x = h ? cc.y : cc.x;
            b.y = h ? 0.0f : cc.z;
            float cnc = cc.w;            // |c_col|^2

            v8f acc0 = {};
            v8f acc1 = {};
            acc0 = __builtin_amdgcn_wmma_f32_16x16x4_f32(false, a0, false, b,
                                                         (short)0, acc0, false, false);
            acc1 = __builtin_amdgcn_wmma_f32_16x16x4_f32(false, a1, false, b,
                                                         (short)0, acc1, false, false);

            // D layout: VGPR v, lanes 0-15 -> row v, lanes 16-31 -> row v+8.
#pragma unroll
            for (int v = 0; v < 8; ++v) {
                T[(v + 8 * h) * RSTRIDE + t * 16 + lo]      = cnc - 2.0f * acc0[v];
                T[(16 + v + 8 * h) * RSTRIDE + t * 16 + lo] = cnc - 2.0f * acc1[v];
            }
        }
        __syncthreads();

        // Each lane scans the 64 candidates of its own row (float4 DS reads).
        const float4* Trow = (const float4*)&T[lane * RSTRIDE];
#pragma unroll 4
        for (int c4 = 0; c4 < 16; ++c4) {
            float4 q = Trow[c4];
            float sv[4] = {q.x, q.y, q.z, q.w};
#pragma unroll
            for (int e = 0; e < 4; ++e) {
                float s = sv[e];
                int j = cbase + c4 * 4 + e;
                if (j != r && s < dist[KNB - 1]) {
                    int p = 0;
#pragma unroll
                    for (int m = 0; m < KNB; ++m) p += (dist[m] <= s) ? 1 : 0;
#pragma unroll
                    for (int m = KNB - 1; m >= 1; --m) {
                        bool mv = (dist[m - 1] > s);
                        dist[m] = mv ? dist[m - 1] : dist[m];
                        idx[m]  = mv ? idx[m - 1]  : idx[m];
                    }
#pragma unroll
                    for (int m = 0; m < KNB; ++m) {
                        if (m == p) { dist[m] = s; idx[m] = j; }
                    }
                }
            }
        }
        __syncthreads();
    }

#pragma unroll
    for (int m = 0; m < KNB; ++m) nn[r * KNB + m] = idx[m];
}

// ---------------------------------------------------------------------------
// Kernel 3: crossing tests. One thread per (face, neighbor); deterministic
// block reduction of prob-weighted crossing counts into per-block partials.
// ---------------------------------------------------------------------------
__global__ __launch_bounds__(256)
void ecl_cross_kernel(const float4* __restrict__ edg,
                      const int* __restrict__ nn,
                      const float* __restrict__ probs,
                      float* __restrict__ partials,
                      int F) {
    __shared__ float sdata[256];
    int t = blockIdx.x * blockDim.x + threadIdx.x;
    float contrib = 0.0f;
    if (t < F * KNB) {
        int f = t / KNB;
        int n = t - f * KNB;
        int g = nn[f * KNB + n];

        float4 ef0 = edg[3 * f + 0], ef1 = edg[3 * f + 1], ef2 = edg[3 * f + 2];
        float4 eg0 = edg[3 * g + 0], eg1 = edg[3 * g + 1], eg2 = edg[3 * g + 2];
        float efx[3] = {ef0.x, ef1.x, ef2.x};
        float efy[3] = {ef0.y, ef1.y, ef2.y};
        float efz[3] = {ef0.z, ef1.z, ef2.z};
        float egx[3] = {eg0.x, eg1.x, eg2.x};
        float egy[3] = {eg0.y, eg1.y, eg2.y};
        float egz[3] = {eg0.z, eg1.z, eg2.z};

        int cnt = 0;
#pragma unroll
        for (int i = 0; i < 3; ++i) {
#pragma unroll
            for (int k = 0; k < 3; ++k) {
                // cr = e_i x nb_k
                float crx = efy[i] * egz[k] - efz[i] * egy[k];
                float cry = efz[i] * egx[k] - efx[i] * egz[k];
                float crz = efx[i] * egy[k] - efy[i] * egx[k];
                float den = crx * efx[i] + cry * efy[i] + crz * efz[i];
                float tn  = crx * egx[k] + cry * egy[k] + crz * egz[k];
                float un  = crx * efx[k] + cry * efy[k] + crz * efz[k];
                float tt = tn / den;
                float uu = un / den;
                cnt += (tt >= 0.0f && tt <= 1.0f && uu >= 0.0f && uu <= 1.0f) ? 1 : 0;
            }
        }
        contrib = probs[f] * (float)cnt;
    }

    sdata[threadIdx.x] = contrib;
    __syncthreads();
#pragma unroll
    for (int s = 128; s > 0; s >>= 1) {
        if ((int)threadIdx.x < s) sdata[threadIdx.x] += sdata[threadIdx.x + s];
        __syncthreads();
    }
    if (threadIdx.x == 0) partials[blockIdx.x] = sdata[0];
}

// ---------------------------------------------------------------------------
// Kernel 4: deterministic final reduction; out = sum / F.
// ---------------------------------------------------------------------------
__global__ __launch_bounds__(256)
void ecl_reduce_kernel(const float* __restrict__ partials, int n,
                       float* __restrict__ out, int F) {
    __shared__ float sdata[256];
    float acc = 0.0f;
    for (int i = threadIdx.x; i < n; i += 256) acc += partials[i];
    sdata[threadIdx.x] = acc;
    __syncthreads();
#pragma unroll
    for (int s = 128; s > 0; s >>= 1) {
        if ((int)threadIdx.x < s) sdata[threadIdx.x] += sdata[threadIdx.x + s];
        __syncthreads();
    }
    if (threadIdx.x == 0) out[0] = sdata[0] / (float)F;
}

// ---------------------------------------------------------------------------
extern "C" void kernel_launch(void* const* d_in, const int* in_sizes, int n_in,
                              void* d_out, int out_size, void* d_ws, size_t ws_size,
                              hipStream_t stream) {
    const float* verts = (const float*)d_in[0];   // V x 3 f32
    const int*   faces = (const int*)d_in[1];     // F x 3 i32
    const float* probs = (const float*)d_in[2];   // F f32
    float* out = (float*)d_out;

    const int F = in_sizes[2];

    // Workspace layout (bytes from d_ws)
    char* ws = (char*)d_ws;
    float4* cent     = (float4*)(ws);                         // F * 16
    float4* edg      = (float4*)(ws + (size_t)F * 16);        // F * 48
    int*    nn       = (int*)   (ws + (size_t)F * 64);        // F * KNB * 4
    float*  partials = (float*) (ws + (size_t)F * 64 + (size_t)F * KNB * 4);

    // K1: centroids + edges
    {
        int blocks = (F + 255) / 256;
        ecl_prep_kernel<<<blocks, 256, 0, stream>>>(verts, faces, cent, edg, F);
    }
    // K2: WMMA k-NN (wave handles 32 rows; 4 waves/block)
    {
        int waves = F / 32;
        int blocks = (waves + 3) / 4;
        ecl_topk_kernel<<<blocks, 128, 0, stream>>>(cent, nn, F);
    }
    // K3: crossing counts + per-block partial sums
    int blocks3 = (F * KNB + 255) / 256;
    ecl_cross_kernel<<<blocks3, 256, 0, stream>>>(edg, nn, probs, partials, F);
    // K4: final deterministic reduction
    ecl_reduce_kernel<<<1, 256, 0, stream>>>(partials, blocks3, out, F);
}